// GCNEncoder_2817498546746
// MI455X (gfx1250) — compile-verified
//
#include <hip/hip_runtime.h>
#include <hip/hip_bf16.h>

// ---------------------------------------------------------------------------
// 2-layer GCN for MI455X (gfx1250).
//   - fp32 WMMA (v_wmma_f32_16x16x4_f32) for the two dense GEMMs (exact fp32)
//   - L2-resident scatter-add aggregation with global_atomic_add_f32
//   - edge norm computed once, shared by both layers (cached=True semantics)
// ---------------------------------------------------------------------------

typedef __attribute__((ext_vector_type(2))) float v2f;
typedef __attribute__((ext_vector_type(8))) float v8f;

static constexpr int IN_CH = 128;
static constexpr int NHID  = 64;
static constexpr int OUT_CH = 32;

// ---------------------------------------------------------------- utilities
__global__ void fill_f32_kernel(float* __restrict__ p, float v, int n) {
    int i = blockIdx.x * blockDim.x + threadIdx.x;
    if (i < n) p[i] = v;
}

// deg[dst[e]] += 1   (deg pre-initialized to 1.0 for the self loop)
__global__ void degree_kernel(const int* __restrict__ dst,
                              float* __restrict__ deg, int E) {
    int e = blockIdx.x * blockDim.x + threadIdx.x;
    if (e < E) atomicAdd(&deg[dst[e]], 1.0f);
}

// in-place deg -> deg^{-1/2}  (deg >= 1 always, so no zero guard needed)
__global__ void rsqrt_kernel(float* __restrict__ d, int n) {
    int i = blockIdx.x * blockDim.x + threadIdx.x;
    if (i < n) d[i] = rsqrtf(d[i]);
}

// norm[e] = dis[src[e]] * dis[dst[e]]
__global__ void norm_kernel(const int* __restrict__ src,
                            const int* __restrict__ dst,
                            const float* __restrict__ dis,
                            float* __restrict__ norm, int E) {
    int e = blockIdx.x * blockDim.x + threadIdx.x;
    if (e < E) norm[e] = dis[src[e]] * dis[dst[e]];
}

// ------------------------------------------------------------- WMMA GEMM
// C[M x NC] = A[M x K] * B[K x NC], row-major, fp32, one 16x16 tile per wave.
// Uses V_WMMA_F32_16X16X4_F32: D(16x16,f32) = A(16x4,f32)*B(4x16,f32) + C.
// A frag (ISA 7.12.2): lanes 0-15 hold M=lane, K={k,k+1}; lanes 16-31 hold
// M=lane-16, K={k+2,k+3}. B frag mirrored (K = k + 2*half + vgpr).
// C/D: VGPR j -> row j (lanes 0-15) / row j+8 (lanes 16-31), col = lane&15.
template <int K, int NC>
__global__ void wmma_gemm_kernel(const float* __restrict__ A,
                                 const float* __restrict__ B,
                                 float* __restrict__ C, int tilesTotal) {
    const int wave = blockIdx.x * (blockDim.x >> 5) + (threadIdx.x >> 5);
    if (wave >= tilesTotal) return;           // wave-uniform: EXEC stays full
    const int lane = threadIdx.x & 31;
    constexpr int NT = NC / 16;
    const int tm = wave / NT;
    const int tn = wave % NT;
    const int half = lane >> 4;               // 0: K lo pair, 1: K hi pair
    const int m    = lane & 15;

    const float* __restrict__ arow = A + (size_t)(tm * 16 + m) * K;
    const int colg = tn * 16 + m;

    v8f acc = {};
#pragma unroll 4
    for (int k = 0; k < K; k += 4) {
        const int ka = k + 2 * half;
        v2f a;
        a.x = arow[ka];
        a.y = arow[ka + 1];
        v2f b;
        b.x = B[(size_t)ka * NC + colg];
        b.y = B[(size_t)(ka + 1) * NC + colg];
        acc = __builtin_amdgcn_wmma_f32_16x16x4_f32(
            /*neg_a=*/false, a, /*neg_b=*/false, b,
            /*c_mod=*/(short)0, acc, /*reuse_a=*/false, /*reuse_b=*/false);
    }

#pragma unroll
    for (int j = 0; j < 8; ++j) {
        C[(size_t)(tm * 16 + j + 8 * half) * NC + colg] = acc[j];
    }
}

// ------------------------------------------------------- edge aggregation
// agg[dst[e]*C + c] += m[src[e]*C + c] * norm[e]
// One thread per (edge, channel); lanes of a wave share an edge -> scalar
// broadcast loads of src/dst/norm, fully coalesced gather of m[src].
template <int C>
__global__ void scatter_kernel(const int* __restrict__ src,
                               const int* __restrict__ dst,
                               const float* __restrict__ norm,
                               const float* __restrict__ m,
                               float* __restrict__ agg, int E) {
    constexpr int SH = (C == 64) ? 6 : 5;
    long long idx = (long long)blockIdx.x * blockDim.x + threadIdx.x;
    int e = (int)(idx >> SH);
    int c = (int)(idx & (C - 1));
    if (e < E) {
        int s = src[e];
        int d = dst[e];
        float v = m[(size_t)s * C + c] * norm[e];
        atomicAdd(&agg[(size_t)d * C + c], v);
    }
}

// ---------------------------------------------- self-loop + bias (+ ReLU)
// out[i,c] = act( agg[i,c] + m[i,c]*dis[i]^2 + bias[c] )
template <int C, bool RELU>
__global__ void finish_kernel(const float* __restrict__ agg,
                              const float* __restrict__ m,
                              const float* __restrict__ dis,
                              const float* __restrict__ bias,
                              float* __restrict__ out, int N) {
    constexpr int SH = (C == 64) ? 6 : 5;
    long long idx = (long long)blockIdx.x * blockDim.x + threadIdx.x;
    int i = (int)(idx >> SH);
    int c = (int)(idx & (C - 1));
    if (i < N) {
        float d = dis[i];
        float v = agg[idx] + m[idx] * (d * d) + bias[c];
        out[idx] = RELU ? fmaxf(v, 0.0f) : v;
    }
}

// ---------------------------------------------------------------- launcher
static inline size_t align256(size_t x) { return (x + 255) & ~(size_t)255; }

extern "C" void kernel_launch(void* const* d_in, const int* in_sizes, int n_in,
                              void* d_out, int out_size, void* d_ws, size_t ws_size,
                              hipStream_t stream) {
    const float* x    = (const float*)d_in[0];
    const int*   eidx = (const int*)d_in[1];   // [2, E] (JAX default -> int32)
    const float* W1   = (const float*)d_in[2];
    const float* b1   = (const float*)d_in[3];
    const float* W2   = (const float*)d_in[4];
    const float* b2   = (const float*)d_in[5];
    float*       out  = (float*)d_out;

    const int N = in_sizes[0] / IN_CH;   // 100000
    const int E = in_sizes[1] / 2;       // 1600000
    const int* src = eidx;
    const int* dst = eidx + E;

    // workspace layout (everything we read is re-initialized each call)
    char* ws = (char*)d_ws;
    size_t off = 0;
    float* deg  = (float*)(ws + off); off = align256(off + (size_t)N * 4);        // -> dis in place
    float* nrm  = (float*)(ws + off); off = align256(off + (size_t)E * 4);
    float* m1   = (float*)(ws + off); off = align256(off + (size_t)N * NHID * 4);
    float* agg1 = (float*)(ws + off); off = align256(off + (size_t)N * NHID * 4); // -> h in place
    float* m2   = (float*)(ws + off); off = align256(off + (size_t)N * OUT_CH * 4);
    float* agg2 = (float*)(ws + off); off = align256(off + (size_t)N * OUT_CH * 4);
    (void)ws_size;

    const int BLK = 256;
    auto blocks = [&](long long n) { return (unsigned)((n + BLK - 1) / BLK); };

    // 0) init: deg = 1 (self loop), agg1 = agg2 = 0
    fill_f32_kernel<<<blocks(N), BLK, 0, stream>>>(deg, 1.0f, N);
    fill_f32_kernel<<<blocks((long long)N * NHID), BLK, 0, stream>>>(agg1, 0.0f, N * NHID);
    fill_f32_kernel<<<blocks((long long)N * OUT_CH), BLK, 0, stream>>>(agg2, 0.0f, N * OUT_CH);

    // 1) degree -> deg^{-1/2} -> edge norm (shared by both layers)
    degree_kernel<<<blocks(E), BLK, 0, stream>>>(dst, deg, E);
    rsqrt_kernel<<<blocks(N), BLK, 0, stream>>>(deg, N);
    norm_kernel<<<blocks(E), BLK, 0, stream>>>(src, dst, deg, nrm, E);
    float* dis = deg;

    // 2) layer 1: m1 = x @ W1  (WMMA fp32), scatter, self-loop+bias+ReLU
    {
        const int tiles = (N / 16) * (NHID / 16);              // 25000
        wmma_gemm_kernel<IN_CH, NHID>
            <<<(tiles + 7) / 8, BLK, 0, stream>>>(x, W1, m1, tiles);
        scatter_kernel<NHID>
            <<<blocks((long long)E * NHID), BLK, 0, stream>>>(src, dst, nrm, m1, agg1, E);
        finish_kernel<NHID, true>
            <<<blocks((long long)N * NHID), BLK, 0, stream>>>(agg1, m1, dis, b1, agg1, N);
    }
    float* h = agg1;

    // 3) layer 2: m2 = h @ W2  (WMMA fp32), scatter, self-loop+bias -> d_out
    {
        const int tiles = (N / 16) * (OUT_CH / 16);            // 12500
        wmma_gemm_kernel<NHID, OUT_CH>
            <<<(tiles + 7) / 8, BLK, 0, stream>>>(h, W2, m2, tiles);
        scatter_kernel<OUT_CH>
            <<<blocks((long long)E * OUT_CH), BLK, 0, stream>>>(src, dst, nrm, m2, agg2, E);
        finish_kernel<OUT_CH, false>
            <<<blocks((long long)N * OUT_CH), BLK, 0, stream>>>(agg2, m2, dis, b2, out, N);
    }
}